// Neigh_Agg_49323404427447
// MI455X (gfx1250) — compile-verified
//
#include <hip/hip_runtime.h>

#define NUM_NODE 100000
#define EMBED_DIM 128
#define N_EDGES 600000

typedef __attribute__((ext_vector_type(2))) float v2f;
typedef __attribute__((ext_vector_type(8))) float v8f;

// ---------------------------------------------------------------------------
// Kernel 1: zero the output accumulator and the degree counters
// ---------------------------------------------------------------------------
__global__ void na_init_kernel(float* __restrict__ out, float* __restrict__ deg) {
    const int idx = blockIdx.x * blockDim.x + threadIdx.x;
    if (idx < NUM_NODE * EMBED_DIM) out[idx] = 0.0f;
    if (idx < NUM_NODE)             deg[idx] = 0.0f;
}

// ---------------------------------------------------------------------------
// Kernel 2: y = relu(x @ W^T + b) via V_WMMA_F32_16X16X4_F32.
// One wave computes a 16x128 output stripe: 8 accumulators (v8f each),
// K-loop over 128 in steps of 4 -> 8*32 = 256 WMMA ops per wave.
//
// A tile (16x4 f32): lanes 0-15 hold row m0+lane with K = {k0, k0+1} in the
// two A VGPRs; lanes 16-31 hold the same rows with K = {k0+2, k0+3}.
// B tile (4x16 f32): lanes 0-15 hold col n0+lane with K = {k0, k0+1};
// lanes 16-31 hold K = {k0+2, k0+3}.  B[k][n] = W[n][k] (W^T product).
// C/D tile: VGPR j -> M = m0 + j + 8*(lane>=16), N = n0 + (lane & 15).
// ---------------------------------------------------------------------------
__global__ void __launch_bounds__(256)
na_gemm_relu_kernel(const float* __restrict__ x,
                    const float* __restrict__ W,
                    const float* __restrict__ b,
                    float* __restrict__ y) {
    const int lane = threadIdx.x & 31;
    const int waveInBlock = threadIdx.x >> 5;
    const int waveId = blockIdx.x * 8 + waveInBlock;
    const int m0 = waveId * 16;
    if (m0 >= NUM_NODE) return;          // uniform per wave -> EXEC all-1s inside

    const int half = lane >> 4;          // 0: K pair {0,1}, 1: K pair {2,3}
    const int l16  = lane & 15;

    const v8f zero = {};
    v8f acc[8];
#pragma unroll
    for (int nt = 0; nt < 8; ++nt) acc[nt] = zero;

    const float* __restrict__ xrow = x + (long long)(m0 + l16) * EMBED_DIM;

    for (int k0 = 0; k0 < EMBED_DIM; k0 += 4) {
        const int kk = k0 + 2 * half;
        v2f a;
        a.x = xrow[kk + 0];
        a.y = xrow[kk + 1];
#pragma unroll
        for (int nt = 0; nt < 8; ++nt) {
            const int n = nt * 16 + l16;
            v2f bm;
            bm.x = W[n * EMBED_DIM + kk + 0];
            bm.y = W[n * EMBED_DIM + kk + 1];
            // 8 args: (neg_a, A, neg_b, B, c_mod, C, reuse_a, reuse_b)
            acc[nt] = __builtin_amdgcn_wmma_f32_16x16x4_f32(
                false, a, false, bm, (short)0, acc[nt], false, false);
        }
    }

    // Epilogue: bias + ReLU, write y
#pragma unroll
    for (int nt = 0; nt < 8; ++nt) {
        const int n = nt * 16 + l16;
        const float bias = b[n];
#pragma unroll
        for (int j = 0; j < 8; ++j) {
            const int m = m0 + half * 8 + j;
            float v = acc[nt][j] + bias;
            y[(long long)m * EMBED_DIM + n] = v > 0.0f ? v : 0.0f;
        }
    }
}

// ---------------------------------------------------------------------------
// Kernel 3: edge scatter.  32 lanes per edge; each lane loads a float4 of
// y[tgt] (coalesced 512B per edge) and issues 4 global f32 atomics into
// out[src].  Both y and out fit in the 192MB L2, so the random traffic is
// on-die.  Lane 0 of each edge bumps the degree counter.
// ---------------------------------------------------------------------------
__global__ void na_scatter_kernel(const int* __restrict__ src,
                                  const int* __restrict__ tgt,
                                  const float* __restrict__ y,
                                  float* __restrict__ out,
                                  float* __restrict__ deg) {
    const int gid = blockIdx.x * blockDim.x + threadIdx.x;
    const int e = gid >> 5;              // edge id
    if (e >= N_EDGES) return;
    const int q  = (gid & 31) * 4;       // feature quad offset

    const int s = src[e];
    const int t = tgt[e];

    const float4 v = *(const float4*)(y + (long long)t * EMBED_DIM + q);
    float* o = out + (long long)s * EMBED_DIM + q;
    atomicAdd(o + 0, v.x);
    atomicAdd(o + 1, v.y);
    atomicAdd(o + 2, v.z);
    atomicAdd(o + 3, v.w);
    if (q == 0) atomicAdd(deg + s, 1.0f);
}

// ---------------------------------------------------------------------------
// Kernel 4: degree normalization out[n][d] /= max(deg[n], 1)
// ---------------------------------------------------------------------------
__global__ void na_norm_kernel(float* __restrict__ out,
                               const float* __restrict__ deg) {
    const int idx = blockIdx.x * blockDim.x + threadIdx.x;
    if (idx >= NUM_NODE * EMBED_DIM) return;
    float d = deg[idx >> 7];
    d = d > 1.0f ? d : 1.0f;
    out[idx] = out[idx] / d;
}

// ---------------------------------------------------------------------------
extern "C" void kernel_launch(void* const* d_in, const int* in_sizes, int n_in,
                              void* d_out, int out_size, void* d_ws, size_t ws_size,
                              hipStream_t stream) {
    const float* x  = (const float*)d_in[0];
    const int*   ei = (const int*)d_in[1];   // [2, N_EDGES]; row0=src, row1=tgt
    const float* W  = (const float*)d_in[2];
    const float* b  = (const float*)d_in[3];
    float* out = (float*)d_out;

    // Workspace layout: deg [NUM_NODE] floats, then y [NUM_NODE*EMBED_DIM] floats
    float* deg = (float*)d_ws;
    float* y   = (float*)((char*)d_ws +
                 (((size_t)NUM_NODE * sizeof(float) + 255) & ~(size_t)255));

    {   // init
        const int total = NUM_NODE * EMBED_DIM;
        na_init_kernel<<<(total + 255) / 256, 256, 0, stream>>>(out, deg);
    }
    {   // GEMM + bias + relu  (6250 wave-tiles, 8 waves / 256-thread block)
        const int waves  = NUM_NODE / 16;            // 6250
        const int blocks = (waves + 7) / 8;          // 782
        na_gemm_relu_kernel<<<blocks, 256, 0, stream>>>(x, W, b, y);
    }
    {   // edge gather + atomic scatter (32 lanes per edge)
        const long long threads = (long long)N_EDGES * 32;
        const int blocks = (int)((threads + 255) / 256);
        na_scatter_kernel<<<blocks, 256, 0, stream>>>(ei, ei + N_EDGES, y, out, deg);
    }
    {   // normalize
        const int total = NUM_NODE * EMBED_DIM;
        na_norm_kernel<<<(total + 255) / 256, 256, 0, stream>>>(out, deg);
    }
}